// CkyLayerABC_14860586844814
// MI455X (gfx1250) — compile-verified
//
#include <hip/hip_runtime.h>
#include <hip/hip_bf16.h>
#include <math.h>

// MI455X (gfx1250, wave32) implementation of:
//   data = logsumexp(emissions[B,N,N,M], axis=-1)   (memory-bound, 256MB @ 23.3TB/s ~= 11us)
//   out  = CKY inside log-partition over data        (latency-bound DP, charts in 320KB LDS)
//
// B=32, N=128, M=128 (fixed per reference).

#define CKY_B 32
#define CKY_N 128
#define CKY_M 128

typedef __attribute__((ext_vector_type(2))) float v2f;
typedef __attribute__((ext_vector_type(8))) float v8f;

// ---------------------------------------------------------------------------
// Kernel 1: logsumexp over labels. One wave handles 16 rows of 128 labels.
// Lane r (0..15) streams labels [0,64) of row r; lane r+16 streams [64,128),
// each with an online (max, expsum). After a shfl_xor(16) row-max merge, each
// lane rescales its partial and places it in its V_WMMA_F32_16X16X4_F32
// A-matrix slot (lanes 0-15 -> K0, lanes 16-31 -> K2; K1/K3 = 0). B = ones,
// so one WMMA produces all 16 row sums (D vgpr j: M=j lanes 0-15, M=j+8
// lanes 16-31). Lanes 0/16 stage sums via LDS; lane r writes max+log(sum).
// ---------------------------------------------------------------------------
__global__ __launch_bounds__(256) void cky_lse_labels(
    const float* __restrict__ em, float* __restrict__ data) {
  const int lane  = threadIdx.x & 31;
  const int wloc  = threadIdx.x >> 5;                 // wave in block (0..7)
  const int r     = lane & 15;                        // row within 16-row tile
  const int half  = lane >> 4;                        // label half (0/1)
  const long wg   = (long)blockIdx.x * 8 + wloc;      // global wave id
  const long row  = wg * 16 + r;                      // row in [0, B*N*N)

  const float4* p = (const float4*)(em + row * (long)CKY_M + half * 64);

  // Online logsumexp over this lane's 64 labels (16 x float4).
  float m = -INFINITY, acc = 0.0f;
#pragma unroll
  for (int i = 0; i < 16; ++i) {
    float4 x = p[i];
    float m4 = fmaxf(fmaxf(x.x, x.y), fmaxf(x.z, x.w));
    float mn = fmaxf(m, m4);
    acc = acc * __expf(m - mn) + ((__expf(x.x - mn) + __expf(x.y - mn)) +
                                  (__expf(x.z - mn) + __expf(x.w - mn)));
    m = mn;
  }

  // Full-row max across the two halves; rescale partial exp-sum.
  float M  = fmaxf(m, __shfl_xor(m, 16));
  float a0 = acc * __expf(m - M);

  // Single f32 WMMA row-sum: D = A x ones.  A layout (32-bit 16x4):
  //   vgpr0 = {lanes 0-15: K=0, lanes 16-31: K=2}, vgpr1 = {K=1 | K=3}.
  v2f A;  A[0] = a0;   A[1] = 0.0f;
  v2f Bm; Bm[0] = 1.0f; Bm[1] = 1.0f;   // all-ones B: layout-independent
  v8f C = {};
  v8f D = __builtin_amdgcn_wmma_f32_16x16x4_f32(
      /*neg_a=*/false, A, /*neg_b=*/false, Bm,
      /*c_mod=*/(short)0, C, /*reuse_a=*/false, /*reuse_b=*/false);

  // D vgpr j: lanes 0-15 hold row j, lanes 16-31 hold row j+8 (any column).
  __shared__ float sums[8][16];
  if (r == 0) {  // lanes 0 and 16 of each wave
#pragma unroll
    for (int j = 0; j < 8; ++j) sums[wloc][half * 8 + j] = D[j];
  }
  __syncthreads();

  if (lane < 16) {
    float s = sums[wloc][lane];
    data[wg * 16 + lane] = M + __logf(s);
  }
}

// ---------------------------------------------------------------------------
// Kernel 2: CKY inside DP, one workgroup per batch element. Both charts live
// in LDS with row stride 136 floats (136 mod 64 = 8): the quad-split access
// t1[(k+4i)*S + s] hits bank (8k + s) mod 64 -> conflict-free across a wave.
// 512 threads: 4 threads per span position split the width-v loop; quad
// shfl_xor reductions do the log-semiring combine (max pass, exp-sum pass).
// ---------------------------------------------------------------------------
#define CKY_S 136                                     // padded LDS row stride

__global__ __launch_bounds__(512) void cky_inside(
    const float* __restrict__ data, float* __restrict__ out) {
  const int b = blockIdx.x;
  const float* db = data + (long)b * CKY_N * CKY_N;

  __shared__ float t1[CKY_N * CKY_S];   // t1[w][s]  : span starting at s, width w
  __shared__ float t2[CKY_N * CKY_S];   // t2[row][e]: span ending at e, width N-1-row

  const int tid = threadIdx.x;

  // width-0 init: t1[0][s] = t2[N-1][s] = data[s][s]
  for (int s = tid; s < CKY_N; s += blockDim.x) {
    float d0 = db[s * CKY_N + s];
    t1[s] = d0;
    t2[(CKY_N - 1) * CKY_S + s] = d0;
  }
  __syncthreads();

  const int s = tid >> 2;      // span start position (0..127)
  const int k = tid & 3;       // quad lane splitting the v loop

  for (int w = 1; w < CKY_N; ++w) {
    if (s < CKY_N - w) {
      const float* p1 = t1 + s;                                 // + v*S
      const float* p2 = t2 + (CKY_N - w) * CKY_S + (s + w);     // + v*S
      // pass 1: max over v (split 4 ways)
      float m = -INFINITY;
      for (int v = k; v < w; v += 4)
        m = fmaxf(m, p1[v * CKY_S] + p2[v * CKY_S]);
      m = fmaxf(m, __shfl_xor(m, 1));
      m = fmaxf(m, __shfl_xor(m, 2));
      // pass 2: sum exp(x - m)
      float acc = 0.0f;
      for (int v = k; v < w; v += 4)
        acc += __expf(p1[v * CKY_S] + p2[v * CKY_S] - m);
      acc += __shfl_xor(acc, 1);
      acc += __shfl_xor(acc, 2);
      if (k == 0) {
        float val = m + __logf(acc) + db[s * CKY_N + (s + w)];
        t1[w * CKY_S + s] = val;                 // new row: not read this step
        t2[(CKY_N - 1 - w) * CKY_S + (s + w)] = val;
      }
    }
    __syncthreads();
  }

  if (tid == 0) out[b] = t1[(CKY_N - 1) * CKY_S + 0];
}

// ---------------------------------------------------------------------------
extern "C" void kernel_launch(void* const* d_in, const int* in_sizes, int n_in,
                              void* d_out, int out_size, void* d_ws, size_t ws_size,
                              hipStream_t stream) {
  (void)in_sizes; (void)n_in; (void)out_size; (void)ws_size;
  const float* emissions = (const float*)d_in[0];
  float* data = (float*)d_ws;          // [B,N,N] intermediate, 2 MB
  float* out  = (float*)d_out;         // [B]

  // B*N*N = 524288 rows; 16 rows/wave, 8 waves/block -> 4096 blocks.
  cky_lse_labels<<<4096, 256, 0, stream>>>(emissions, data);
  cky_inside<<<CKY_B, 512, 0, stream>>>(data, out);
}